// DeepLabCE_8194797600924
// MI455X (gfx1250) — compile-verified
//
#include <hip/hip_runtime.h>
#include <hip/hip_bf16.h>
#include <math.h>

// Problem constants (fixed by setup_inputs): B=4, C=19, H=512, W=1024
#define CHN   19
#define HW_   (512 * 1024)          // 524288 = 2^19
#define NPIX  (4 * HW_)             // 2097152 = 2^21
#define CHW   ((size_t)CHN * HW_)
#define KSEL  1468006u              // int(0.7 * NPIX)
#define NBLK  1024                  // blocks for the big passes (256 thr, 8 px/thr)

// ---- gfx1250 async global->LDS path (with inline-asm fallback) -------------
#if defined(__has_builtin)
#  if __has_builtin(__builtin_amdgcn_global_load_async_to_lds_b32)
#    define HAVE_ASYNC_B32 1
#  endif
#  if __has_builtin(__builtin_amdgcn_s_wait_asynccnt)
#    define HAVE_WAIT_ASYNC 1
#  endif
#endif

typedef __attribute__((address_space(1))) int* as1_int_p;
typedef __attribute__((address_space(3))) int* as3_int_p;

__device__ __forceinline__ void async_load_f32_to_lds(const float* gptr, float* lptr) {
#if defined(HAVE_ASYNC_B32)
  __builtin_amdgcn_global_load_async_to_lds_b32(
      (as1_int_p)(void*)gptr, (as3_int_p)(void*)lptr, 0, 0);
#else
  unsigned lds_off = (unsigned)(unsigned long long)lptr;       // low 32 bits = LDS offset
  unsigned long long ga = (unsigned long long)(const void*)gptr;
  asm volatile("global_load_async_to_lds_b32 %0, %1, off"
               :: "v"(lds_off), "v"(ga) : "memory");
#endif
}

__device__ __forceinline__ void wait_asynccnt0() {
#if defined(HAVE_WAIT_ASYNC)
  __builtin_amdgcn_s_wait_asynccnt(0);
#else
  asm volatile("s_wait_asynccnt 0" ::: "memory");
#endif
}

// ---- order-preserving float <-> uint key ----------------------------------
__device__ __forceinline__ unsigned ordkey(float f) {
  unsigned u = __float_as_uint(f);
  return (u & 0x80000000u) ? ~u : (u | 0x80000000u);
}
__device__ __forceinline__ float keyval(unsigned u) {
  unsigned b = (u & 0x80000000u) ? (u & 0x7fffffffu) : ~u;
  return __uint_as_float(b);
}

// ---- kernel 1 (defined first so it leads the disasm): per-pixel loss +
//      top-8-bit histogram (the single HBM-heavy pass) -----------------------
__global__ __launch_bounds__(256) void loss_hist_kernel(
    const float* __restrict__ logits, const float* __restrict__ smooth,
    const float* __restrict__ weight2, float* __restrict__ losses,
    unsigned* __restrict__ ghist0) {
  __shared__ float sw[CHN];
  __shared__ unsigned lh[256];
  lh[threadIdx.x] = 0u;

  // Stage the 19-entry weight vector via the gfx1250 async global->LDS path.
  if (threadIdx.x < CHN)
    async_load_f32_to_lds(weight2 + threadIdx.x, &sw[threadIdx.x]);
  wait_asynccnt0();
  __syncthreads();

  float wr[CHN];
#pragma unroll
  for (int c = 0; c < CHN; ++c) wr[c] = sw[c];

  const int tid = blockIdx.x * 256 + threadIdx.x;
  const int nthreads = NBLK * 256;                 // 262144; NPIX = 8 * nthreads

#pragma unroll 1
  for (int i = 0; i < 8; ++i) {
    const int p  = tid + i * nthreads;
    const int b  = p >> 19;
    const int hw = p & (HW_ - 1);
    const float* lp = logits + (size_t)b * CHW + hw;
    const float* sp = smooth + (size_t)b * CHW + hw;

    float x[CHN];
    float m = -INFINITY;
#pragma unroll
    for (int c = 0; c < CHN; ++c) { x[c] = lp[(size_t)c * HW_]; m = fmaxf(m, x[c]); }
    float se = 0.f;
#pragma unroll
    for (int c = 0; c < CHN; ++c) se += __expf(x[c] - m);
    const float lse = m + __logf(se);

    float acc = 0.f;
#pragma unroll
    for (int c = 0; c < CHN; ++c) {
      const float s = sp[(size_t)c * HW_];
      acc = fmaf(s * wr[c], lse - x[c], acc);      // = -(sum s*w*logp)
    }
    losses[p] = acc;
    atomicAdd(&lh[ordkey(acc) >> 24], 1u);
  }
  __syncthreads();
  const unsigned v = lh[threadIdx.x];
  if (v) atomicAdd(&ghist0[threadIdx.x], v);
}

// ---- kernel 0: init scratch (hists + selection state) ----------------------
__global__ __launch_bounds__(256) void init_kernel(unsigned* ghist, unsigned* sel) {
  for (int i = threadIdx.x; i < 4 * 256; i += 256) ghist[i] = 0u;
  if (threadIdx.x == 0) { sel[0] = 0u; sel[1] = KSEL; }
}

// ---- kernel 2: refine histogram at a given 8-bit digit (L2-resident) -------
__global__ __launch_bounds__(256) void hist_pass_kernel(
    const float* __restrict__ losses, const unsigned* __restrict__ sel,
    unsigned* __restrict__ ghist, int shift) {
  __shared__ unsigned lh[256];
  lh[threadIdx.x] = 0u;
  __syncthreads();
  const unsigned prefix = sel[0];
  const unsigned mask = ~((1u << (shift + 8)) - 1u);   // shift in {16,8,0}
  const int stride = gridDim.x * blockDim.x;
  for (int p = blockIdx.x * blockDim.x + threadIdx.x; p < NPIX; p += stride) {
    const unsigned key = ordkey(losses[p]);
    if ((key & mask) == prefix) atomicAdd(&lh[(key >> shift) & 0xFFu], 1u);
  }
  __syncthreads();
  const unsigned v = lh[threadIdx.x];
  if (v) atomicAdd(&ghist[threadIdx.x], v);
}

// ---- kernel 3: pick the digit containing the k-th largest ------------------
__global__ __launch_bounds__(256) void select_digit_kernel(
    const unsigned* __restrict__ ghist, unsigned* __restrict__ sel, int shift) {
  __shared__ unsigned suf[256];
  const int t = threadIdx.x;
  const unsigned krem = sel[1];
  suf[t] = ghist[t];
  __syncthreads();
  // Hillis-Steele suffix sum: suf[t] = sum_{j>=t} hist[j]
  for (int off = 1; off < 256; off <<= 1) {
    const unsigned add = (t + off < 256) ? suf[t + off] : 0u;
    __syncthreads();
    suf[t] += add;
    __syncthreads();
  }
  const unsigned above = (t < 255) ? suf[t + 1] : 0u;
  if (above < krem && krem <= suf[t]) {            // exactly one winner
    sel[0] |= ((unsigned)t) << shift;
    sel[1] = krem - above;
  }
}

// ---- kernel 4: deterministic partial sums of values above threshold --------
__global__ __launch_bounds__(256) void topk_sum_kernel(
    const float* __restrict__ losses, const unsigned* __restrict__ sel,
    double* __restrict__ psum, unsigned* __restrict__ pcnt) {
  const unsigned T = sel[0];
  double s = 0.0;
  unsigned c = 0u;
  const int stride = gridDim.x * blockDim.x;
  for (int p = blockIdx.x * blockDim.x + threadIdx.x; p < NPIX; p += stride) {
    const float v = losses[p];
    if (ordkey(v) > T) { s += (double)v; c += 1u; }
  }
  __shared__ double ds[256];
  __shared__ unsigned cs[256];
  ds[threadIdx.x] = s;
  cs[threadIdx.x] = c;
  __syncthreads();
  for (int off = 128; off > 0; off >>= 1) {
    if (threadIdx.x < off) {
      ds[threadIdx.x] += ds[threadIdx.x + off];
      cs[threadIdx.x] += cs[threadIdx.x + off];
    }
    __syncthreads();
  }
  if (threadIdx.x == 0) { psum[blockIdx.x] = ds[0]; pcnt[blockIdx.x] = cs[0]; }
}

// ---- kernel 5: finalize (ties at threshold + mean) -------------------------
__global__ __launch_bounds__(256) void finalize_kernel(
    const unsigned* __restrict__ sel, const double* __restrict__ psum,
    const unsigned* __restrict__ pcnt, float* __restrict__ out) {
  __shared__ double ds[256];
  __shared__ unsigned cs[256];
  double s = 0.0;
  unsigned c = 0u;
  for (int i = threadIdx.x; i < NBLK; i += 256) { s += psum[i]; c += pcnt[i]; }
  ds[threadIdx.x] = s;
  cs[threadIdx.x] = c;
  __syncthreads();
  for (int off = 128; off > 0; off >>= 1) {
    if (threadIdx.x < off) {
      ds[threadIdx.x] += ds[threadIdx.x + off];
      cs[threadIdx.x] += cs[threadIdx.x + off];
    }
    __syncthreads();
  }
  if (threadIdx.x == 0) {
    const double tv = (double)keyval(sel[0]);
    const double total = ds[0] + (double)(KSEL - cs[0]) * tv;
    out[0] = (float)(total / (double)KSEL);
  }
}

// ---- launcher --------------------------------------------------------------
extern "C" void kernel_launch(void* const* d_in, const int* in_sizes, int n_in,
                              void* d_out, int out_size, void* d_ws, size_t ws_size,
                              hipStream_t stream) {
  const float* logits  = (const float*)d_in[0];
  // d_in[1] = labels (int64) — unused by the reference computation
  const float* smooth  = (const float*)d_in[2];
  const float* weight2 = (const float*)d_in[3];
  float* out = (float*)d_out;

  char* ws = (char*)d_ws;
  float*    losses = (float*)ws;                                   // 8 MB
  unsigned* ghist  = (unsigned*)(ws + (size_t)NPIX * 4);           // 4 x 256 u32
  unsigned* sel    = (unsigned*)(ws + (size_t)NPIX * 4 + 4096);    // 4 u32 (padded to 64B)
  double*   psum   = (double*)(ws + (size_t)NPIX * 4 + 4096 + 64); // NBLK doubles
  unsigned* pcnt   = (unsigned*)(ws + (size_t)NPIX * 4 + 4096 + 64 + (size_t)NBLK * 8);

  init_kernel<<<1, 256, 0, stream>>>(ghist, sel);
  loss_hist_kernel<<<NBLK, 256, 0, stream>>>(logits, smooth, weight2, losses, ghist);
  select_digit_kernel<<<1, 256, 0, stream>>>(ghist, sel, 24);
  hist_pass_kernel<<<NBLK, 256, 0, stream>>>(losses, sel, ghist + 256, 16);
  select_digit_kernel<<<1, 256, 0, stream>>>(ghist + 256, sel, 16);
  hist_pass_kernel<<<NBLK, 256, 0, stream>>>(losses, sel, ghist + 512, 8);
  select_digit_kernel<<<1, 256, 0, stream>>>(ghist + 512, sel, 8);
  hist_pass_kernel<<<NBLK, 256, 0, stream>>>(losses, sel, ghist + 768, 0);
  select_digit_kernel<<<1, 256, 0, stream>>>(ghist + 768, sel, 0);
  topk_sum_kernel<<<NBLK, 256, 0, stream>>>(losses, sel, psum, pcnt);
  finalize_kernel<<<1, 256, 0, stream>>>(sel, psum, pcnt, out);
}